// FixedAbsolutePositionalEmbedding_4724464025787
// MI455X (gfx1250) — compile-verified
//
#include <hip/hip_runtime.h>

// out[b,s,d] = x[b,s,d] + P[s,d]
//   P[s,d] = sin(s / 10000^(j/D))  for even d (j = d)
//          = cos(s / 10000^(j/D))  for odd  d (j = d-1)
//
// Shapes fixed by the reference: B=8, S=4096, D=1024 (fp32).
// Memory-bound: 256 MiB HBM traffic -> ~11.5 us floor at 23.3 TB/s.

typedef float v4f __attribute__((ext_vector_type(4)));

__global__ __launch_bounds__(256) void
posemb_add_kernel(const float* __restrict__ x, float* __restrict__ out) {
    constexpr int   S  = 4096;
    constexpr int   D  = 1024;
    constexpr int   B  = 8;
    constexpr int   D4 = D / 4;                         // 256 float4 per row
    constexpr float INV_2PI = 0.15915494309189535f;     // 1/(2*pi)
    // -log2(10000)/D : per-column exponent slope for 10000^(-j/D) = exp2(c*j)
    constexpr float C = -13.287712379549449f / (float)D;

    const int tid = blockIdx.x * blockDim.x + threadIdx.x; // 0 .. S*D4-1
    const int d4  = tid & (D4 - 1);
    const int s   = tid >> 8;                              // tid / D4
    const int d   = d4 << 2;                               // first of 4 columns

    // Frequencies (in "turns", i.e. pre-divided by 2*pi) for columns d and d+2.
    // v_exp_f32 x2 — computed once, reused across all 8 batches.
    const float f0 = __builtin_amdgcn_exp2f(C * (float)d)       * INV_2PI;
    const float f1 = __builtin_amdgcn_exp2f(C * (float)(d + 2)) * INV_2PI;

    // Range-reduce into [0,1) turns, then use the hw trig pipe directly:
    // V_SIN_F32 / V_COS_F32 compute sin/cos(2*pi * input).
    float t0 = (float)s * f0;  t0 -= __builtin_floorf(t0);
    float t1 = (float)s * f1;  t1 -= __builtin_floorf(t1);

    v4f emb;
    emb.x = __builtin_amdgcn_sinf(t0);   // even col d   -> sin
    emb.y = __builtin_amdgcn_cosf(t0);   // odd  col d+1 -> cos (same angle)
    emb.z = __builtin_amdgcn_sinf(t1);   // even col d+2 -> sin
    emb.w = __builtin_amdgcn_cosf(t1);   // odd  col d+3 -> cos

    const size_t base         = (size_t)tid * 4;     // element offset in one image
    constexpr size_t IMG_ELEM = (size_t)S * D;       // 4M elements per batch image

    // Issue all 8 non-temporal b128 loads first (32 VGPRs in flight),
    // then add + non-temporal b128 stores. Touch-once data: TH=NT.
    v4f v[B];
#pragma unroll
    for (int b = 0; b < B; ++b) {
        const v4f* src =
            reinterpret_cast<const v4f*>(x + (size_t)b * IMG_ELEM + base);
        v[b] = __builtin_nontemporal_load(src);
    }
#pragma unroll
    for (int b = 0; b < B; ++b) {
        v4f r = v[b] + emb;
        __builtin_nontemporal_store(
            r, reinterpret_cast<v4f*>(out + (size_t)b * IMG_ELEM + base));
    }
}

extern "C" void kernel_launch(void* const* d_in, const int* in_sizes, int n_in,
                              void* d_out, int out_size, void* d_ws, size_t ws_size,
                              hipStream_t stream) {
    (void)in_sizes; (void)n_in; (void)out_size; (void)d_ws; (void)ws_size;
    const float* x   = (const float*)d_in[0];
    float*       out = (float*)d_out;

    constexpr int S  = 4096;
    constexpr int D4 = 1024 / 4;
    constexpr int THREADS = 256;                       // 8 wave32 per block
    constexpr int BLOCKS  = (S * D4) / THREADS;        // 4096 blocks

    posemb_add_kernel<<<BLOCKS, THREADS, 0, stream>>>(x, out);
}